// StaticKGEncoder_33938831573719
// MI455X (gfx1250) — compile-verified
//
#include <hip/hip_runtime.h>
#include <math.h>

typedef __attribute__((ext_vector_type(2))) float v2f;
typedef __attribute__((ext_vector_type(8))) float v8f;

#define H_    8
#define D_    32
#define HID_  256
#define NU    20000
#define NL    50000
#define NE    300000
#define FIN   128

// ---------- helpers ----------
__device__ __forceinline__ unsigned fenc(float f) {
    unsigned u = __float_as_uint(f);
    return (u & 0x80000000u) ? ~u : (u | 0x80000000u);
}
__device__ __forceinline__ float fdec(unsigned u) {
    return __uint_as_float((u & 0x80000000u) ? (u & 0x7FFFFFFFu) : ~u);
}

__global__ void fill_f32(float* p, float v, long n) {
    long i = (long)blockIdx.x * blockDim.x + threadIdx.x;
    if (i < n) p[i] = v;
}
__global__ void fill_u32(unsigned* p, unsigned v, long n) {
    long i = (long)blockIdx.x * blockDim.x + threadIdx.x;
    if (i < n) p[i] = v;
}

// Batched transpose: out[z][n][k] = in[z][k][n]   (weights are tiny; one-time cost)
__global__ void transpose_k(const float* __restrict__ in, float* __restrict__ out,
                            int K, int N) {
    long base = (long)blockIdx.z * K * N;
    int idx = blockIdx.x * blockDim.x + threadIdx.x;
    if (idx >= K * N) return;
    int k = idx / N, n = idx - k * N;
    out[base + (size_t)n * K + k] = in[base + idx];
}

// ---------- WMMA fp32 GEMM: C[M,N] = A[M,K] @ W[K,N] (+bias, epilogue) ----------
// W supplied TRANSPOSED as WT[N,K]: A and B fragments are contiguous b64 loads.
// Each wave computes a 32 x (NT_*16) strip (two M-tiles share B fragments) with
// software-pipelined (double-buffered) fragment loads feeding v_wmma_f32_16x16x4_f32.
// The second M-tile of the last pair may not exist (M % 32 == 16): its loads alias
// tile 0 (no OOB) and only its stores are predicated (wave-uniform -> EXEC all-1s).
// blockIdx.y = head batch (per-head offsets aHead/wHead/cHead).
// mode 0: +bias   mode 1: +bias, ReLU   mode 2: +bias, sig*val + (1-sig)*skipsrc
template <int NT_>
__global__ void gemm_wmma_f32(const float* __restrict__ A, int lda,
                              const float* __restrict__ WT,
                              const float* __restrict__ bias,
                              float* __restrict__ C, int ldc,
                              int M, int K, int Nstrips,
                              int mode, const float* __restrict__ skip,
                              const float* __restrict__ skipsrc, int ldskip,
                              int aHead, int wHead, int cHead) {
    int hh = blockIdx.y;
    A  += (size_t)hh * aHead;
    WT += (size_t)hh * wHead;
    C  += (size_t)hh * cHead;

    int wave = (int)((blockIdx.x * (long)blockDim.x + threadIdx.x) >> 5);
    int lane = threadIdx.x & 31;
    int Mpairs = (M + 31) >> 5;
    int mt = wave % Mpairs;
    int ns = wave / Mpairs;
    if (ns >= Nstrips) return;          // wave-uniform exit

    int lhalf = lane >> 4;              // 0: K pair {0,1}; 1: K pair {2,3}
    int l16   = lane & 15;
    bool two  = ((mt << 5) + 16) < M;   // second 16-row tile exists (wave-uniform)
    int row0  = (mt << 5) + l16;
    int row2  = row0 + (two ? 16 : 0);  // alias tile 0 when absent: no OOB loads
    int ncol0 = ((ns * NT_) << 4) + l16;

    const float* Ar0 = A + (size_t)row0 * lda;
    const float* Ar2 = A + (size_t)row2 * lda;
    const float* Wp[NT_];
#pragma unroll
    for (int j = 0; j < NT_; ++j) Wp[j] = WT + (size_t)(ncol0 + (j << 4)) * K;

    v8f acc0[NT_] = {};
    v8f acc2[NT_] = {};

    int ka = lhalf << 1;
    v2f a0 = *reinterpret_cast<const v2f*>(Ar0 + ka);
    v2f a2 = *reinterpret_cast<const v2f*>(Ar2 + ka);
    v2f b[NT_];
#pragma unroll
    for (int j = 0; j < NT_; ++j) b[j] = *reinterpret_cast<const v2f*>(Wp[j] + ka);

    for (int k = 0; k < K - 4; k += 4) {
        int kn = k + 4 + (lhalf << 1);
        v2f a0n = *reinterpret_cast<const v2f*>(Ar0 + kn);
        v2f a2n = *reinterpret_cast<const v2f*>(Ar2 + kn);
        v2f bn[NT_];
#pragma unroll
        for (int j = 0; j < NT_; ++j) bn[j] = *reinterpret_cast<const v2f*>(Wp[j] + kn);
#pragma unroll
        for (int j = 0; j < NT_; ++j)
            acc0[j] = __builtin_amdgcn_wmma_f32_16x16x4_f32(
                false, a0, false, b[j], (short)0, acc0[j], false, false);
#pragma unroll
        for (int j = 0; j < NT_; ++j)
            acc2[j] = __builtin_amdgcn_wmma_f32_16x16x4_f32(
                false, a2, false, b[j], (short)0, acc2[j], false, false);
        a0 = a0n; a2 = a2n;
#pragma unroll
        for (int j = 0; j < NT_; ++j) b[j] = bn[j];
    }
#pragma unroll
    for (int j = 0; j < NT_; ++j)
        acc0[j] = __builtin_amdgcn_wmma_f32_16x16x4_f32(
            false, a0, false, b[j], (short)0, acc0[j], false, false);
#pragma unroll
    for (int j = 0; j < NT_; ++j)
        acc2[j] = __builtin_amdgcn_wmma_f32_16x16x4_f32(
            false, a2, false, b[j], (short)0, acc2[j], false, false);

    float sg = 0.0f;
    if (mode == 2) sg = 1.0f / (1.0f + expf(-skip[0]));

#pragma unroll
    for (int j = 0; j < NT_; ++j) {
        int col = ncol0 + (j << 4);
        float bv = bias ? bias[col] : 0.0f;
#pragma unroll
        for (int r = 0; r < 8; ++r) {
            int rr = (mt << 5) + r + (lhalf << 3);   // C layout: VGPR r -> M = r + 8*lhalf
            float val = acc0[j][r] + bv;
            if (mode == 1) val = fmaxf(val, 0.0f);
            else if (mode == 2)
                val = sg * val + (1.0f - sg) * skipsrc[(size_t)rr * ldskip + col];
            C[(size_t)rr * ldc + col] = val;
        }
        if (two) {
#pragma unroll
            for (int r = 0; r < 8; ++r) {
                int rr = (mt << 5) + 16 + r + (lhalf << 3);
                float val = acc2[j][r] + bv;
                if (mode == 1) val = fmaxf(val, 0.0f);
                else if (mode == 2)
                    val = sg * val + (1.0f - sg) * skipsrc[(size_t)rr * ldskip + col];
                C[(size_t)rr * ldc + col] = val;
            }
        }
    }
}

// ---------- edge kernels ----------
__global__ void edge_alpha(const int* __restrict__ src, const int* __restrict__ dst,
                           const float* __restrict__ q, const float* __restrict__ krel,
                           const float* __restrict__ p_rel, float* __restrict__ alpha,
                           unsigned* __restrict__ amaxE, int E) {
    long idx = (long)blockIdx.x * blockDim.x + threadIdx.x;
    if (idx >= (long)E * H_) return;
    int e = (int)(idx >> 3), h = (int)(idx & 7);
    int s = src[e], d = dst[e];
    const float4* qr = (const float4*)(q    + (size_t)d * HID_ + h * D_);
    const float4* kr = (const float4*)(krel + (size_t)s * HID_ + h * D_);
    float acc = 0.0f;
#pragma unroll
    for (int i = 0; i < 8; ++i) {
        float4 a = qr[i], b = kr[i];
        acc += a.x * b.x + a.y * b.y + a.z * b.z + a.w * b.w;
    }
    float av = acc * p_rel[h] * 0.17677669529663687f;   // 1/sqrt(32)
    alpha[idx] = av;
    atomicMax(&amaxE[(size_t)d * H_ + h], fenc(av));
}

__global__ void edge_exp(const int* __restrict__ dst, float* __restrict__ alpha,
                         const unsigned* __restrict__ amaxE, float* __restrict__ den, int E) {
    long idx = (long)blockIdx.x * blockDim.x + threadIdx.x;
    if (idx >= (long)E * H_) return;
    int e = (int)(idx >> 3), h = (int)(idx & 7);
    int d = dst[e];
    float w = expf(alpha[idx] - fdec(amaxE[(size_t)d * H_ + h]));
    alpha[idx] = w;
    atomicAdd(&den[(size_t)d * H_ + h], w);
}

__global__ void edge_scatter(const int* __restrict__ src, const int* __restrict__ dst,
                             const float* __restrict__ alpha, const float* __restrict__ den,
                             const float* __restrict__ vrel, float* __restrict__ out, int E) {
    long idx = (long)blockIdx.x * blockDim.x + threadIdx.x;
    if (idx >= (long)E * HID_) return;
    int e = (int)(idx >> 8);
    int c = (int)(idx & 255);
    int h = c >> 5;
    int s = src[e], d = dst[e];
    float attn = alpha[(size_t)e * H_ + h] / den[(size_t)d * H_ + h];
    atomicAdd(&out[(size_t)d * HID_ + c], vrel[(size_t)s * HID_ + c] * attn);
}

__global__ void gelu_kernel(const float* __restrict__ x, float* __restrict__ y, long n) {
    long i = (long)blockIdx.x * blockDim.x + threadIdx.x;
    if (i >= n) return;
    float v = x[i];
    y[i] = 0.5f * v * (1.0f + erff(v * 0.70710678118654752f));   // exact GELU
}

// ---------- host orchestration ----------
static void launch_gemm(int NT, const float* A, int lda, const float* WT,
                        const float* bias, float* C, int ldc, int M, int K, int N,
                        int mode, const float* skip, const float* skipsrc, int ldskip,
                        int heads, int aHead, int wHead, int cHead, hipStream_t s) {
    int Mpairs  = (M + 31) / 32;
    int Nstrips = N / (NT * 16);
    int waves   = Mpairs * Nstrips;
    dim3 grid((waves + 3) / 4, heads);  // 128 threads = 4 waves per block
    if (NT == 4)
        gemm_wmma_f32<4><<<grid, 128, 0, s>>>(A, lda, WT, bias, C, ldc, M, K, Nstrips,
                                              mode, skip, skipsrc, ldskip,
                                              aHead, wHead, cHead);
    else
        gemm_wmma_f32<2><<<grid, 128, 0, s>>>(A, lda, WT, bias, C, ldc, M, K, Nstrips,
                                              mode, skip, skipsrc, ldskip,
                                              aHead, wHead, cHead);
}

static void launch_transpose(const float* W, float* WT, int K, int N, int batch,
                             hipStream_t s) {
    int elems = K * N;
    dim3 grid((elems + 255) / 256, 1, batch);
    transpose_k<<<grid, 256, 0, s>>>(W, WT, K, N);
}

extern "C" void kernel_launch(void* const* d_in, const int* in_sizes, int n_in,
                              void* d_out, int out_size, void* d_ws, size_t ws_size,
                              hipStream_t stream) {
    (void)in_sizes; (void)n_in; (void)out_size; (void)ws_size;

    const float* x_user = (const float*)d_in[0];
    const float* x_loc  = (const float*)d_in[1];
    const int* e_src[2] = {(const int*)d_in[2], (const int*)d_in[4]};
    const int* e_dst[2] = {(const int*)d_in[3], (const int*)d_in[5]};
    auto P = [&](int i) { return (const float*)d_in[6 + i]; };
    // params (insertion order of make_params), relative to d_in[6]:
    //  0..3 : in_w_user, in_b_user, in_w_location, in_b_location
    //  layer l base lb = 4 + l*24
    //    type t base tb = lb + t*9 : k_w,k_b,q_w,q_b,v_w,v_b,o_w,o_b,skip
    //    rel ei base rb = lb + 18 + ei*3 : a_rel, m_rel, p_rel
    //  52..55 : lin1_w, lin1_b, lin2_w, lin2_b

    int Nn[2] = {NU, NL};

    float* ws = (float*)d_ws;
    size_t off = 0;
    auto alloc = [&](size_t n) { float* p = ws + off; off += n; return p; };
    float* xdA[2]  = {alloc((size_t)NU * HID_), alloc((size_t)NL * HID_)};
    float* xdB[2]  = {alloc((size_t)NU * HID_), alloc((size_t)NL * HID_)};
    float* kb[2]   = {alloc((size_t)NU * HID_), alloc((size_t)NL * HID_)};
    float* qb[2]   = {alloc((size_t)NU * HID_), alloc((size_t)NL * HID_)};
    float* vb[2]   = {alloc((size_t)NU * HID_), alloc((size_t)NL * HID_)};
    float* krel    = alloc((size_t)NL * HID_);
    float* vrel    = alloc((size_t)NL * HID_);
    float* alpha   = alloc((size_t)NE * H_);
    unsigned* amaxE = (unsigned*)alloc((size_t)NL * H_);
    float* den     = alloc((size_t)NL * H_);
    float* outb[2] = {alloc((size_t)NU * HID_), alloc((size_t)NL * HID_)};
    float* gtmp    = krel;   // krel is dead by the time GELU runs; alias it

    // transposed weights [N,K] (one-time, tiny)
    float* wtin[2]      = {alloc(FIN * HID_), alloc(FIN * HID_)};
    float* wt_main[16];                 // [l][t][k,q,v,o]
    for (int i = 0; i < 16; ++i) wt_main[i] = alloc(HID_ * HID_);
    float* wt_rel[8];                   // [l][ei][a,m]  (each H*D*D)
    for (int i = 0; i < 8; ++i) wt_rel[i] = alloc(H_ * D_ * D_);
    float* wt_lin[2]    = {alloc(HID_ * HID_), alloc(HID_ * HID_)};

    // ---- transpose all weights ----
    launch_transpose(P(0), wtin[0], FIN, HID_, 1, stream);
    launch_transpose(P(2), wtin[1], FIN, HID_, 1, stream);
    for (int l = 0; l < 2; ++l) {
        int lb = 4 + l * 24;
        for (int t = 0; t < 2; ++t) {
            int tb = lb + t * 9;
            for (int m = 0; m < 4; ++m)   // k,q,v,o weights at tb+0,2,4,6
                launch_transpose(P(tb + 2 * m), wt_main[(l * 2 + t) * 4 + m],
                                 HID_, HID_, 1, stream);
        }
        for (int ei = 0; ei < 2; ++ei) {
            int rb = lb + 18 + ei * 3;
            launch_transpose(P(rb + 0), wt_rel[(l * 2 + ei) * 2 + 0], D_, D_, H_, stream);
            launch_transpose(P(rb + 1), wt_rel[(l * 2 + ei) * 2 + 1], D_, D_, H_, stream);
        }
    }
    launch_transpose(P(52), wt_lin[0], HID_, HID_, 1, stream);
    launch_transpose(P(54), wt_lin[1], HID_, HID_, 1, stream);

    // ---- input projections (+ReLU) ----
    launch_gemm(4, x_user, FIN, wtin[0], P(1), xdA[0], HID_, NU, FIN, HID_, 1,
                nullptr, nullptr, 0, 1, 0, 0, 0, stream);
    launch_gemm(4, x_loc,  FIN, wtin[1], P(3), xdA[1], HID_, NL, FIN, HID_, 1,
                nullptr, nullptr, 0, 1, 0, 0, 0, stream);

    float* cur[2] = {xdA[0], xdA[1]};
    float* nxt[2] = {xdB[0], xdB[1]};
    const int ST[2] = {0, 1}, DT[2] = {1, 0};   // ET = [(user,loc),(loc,user)]

    for (int l = 0; l < 2; ++l) {
        int lb = 4 + l * 24;
        for (int t = 0; t < 2; ++t) {
            int tb = lb + t * 9;
            float** wt = &wt_main[(l * 2 + t) * 4];
            launch_gemm(4, cur[t], HID_, wt[0], P(tb + 1), kb[t], HID_,
                        Nn[t], HID_, HID_, 0, nullptr, nullptr, 0, 1, 0, 0, 0, stream);
            launch_gemm(4, cur[t], HID_, wt[1], P(tb + 3), qb[t], HID_,
                        Nn[t], HID_, HID_, 0, nullptr, nullptr, 0, 1, 0, 0, 0, stream);
            launch_gemm(4, cur[t], HID_, wt[2], P(tb + 5), vb[t], HID_,
                        Nn[t], HID_, HID_, 0, nullptr, nullptr, 0, 1, 0, 0, 0, stream);
        }
        {   // zero attention accumulators
            long n0 = (long)NU * HID_, n1 = (long)NL * HID_;
            fill_f32<<<(unsigned)((n0 + 255) / 256), 256, 0, stream>>>(outb[0], 0.0f, n0);
            fill_f32<<<(unsigned)((n1 + 255) / 256), 256, 0, stream>>>(outb[1], 0.0f, n1);
        }
        for (int ei = 0; ei < 2; ++ei) {
            int st = ST[ei], dt = DT[ei];
            int rb = lb + 18 + ei * 3;
            const float* p_rel = P(rb + 2);
            // per-head 32x32 relation transforms, head-batched via blockIdx.y
            launch_gemm(2, kb[st], HID_, wt_rel[(l * 2 + ei) * 2 + 0], nullptr,
                        krel, HID_, Nn[st], D_, D_, 0, nullptr, nullptr, 0,
                        H_, D_, D_ * D_, D_, stream);
            launch_gemm(2, vb[st], HID_, wt_rel[(l * 2 + ei) * 2 + 1], nullptr,
                        vrel, HID_, Nn[st], D_, D_, 0, nullptr, nullptr, 0,
                        H_, D_, D_ * D_, D_, stream);
            long nm = (long)Nn[dt] * H_;
            fill_u32<<<(unsigned)((nm + 255) / 256), 256, 0, stream>>>(amaxE, 0u, nm);
            fill_f32<<<(unsigned)((nm + 255) / 256), 256, 0, stream>>>(den, 0.0f, nm);
            long na = (long)NE * H_;
            edge_alpha<<<(unsigned)((na + 255) / 256), 256, 0, stream>>>(
                e_src[ei], e_dst[ei], qb[dt], krel, p_rel, alpha, amaxE, NE);
            edge_exp<<<(unsigned)((na + 255) / 256), 256, 0, stream>>>(
                e_dst[ei], alpha, amaxE, den, NE);
            long nsct = (long)NE * HID_;
            edge_scatter<<<(unsigned)((nsct + 255) / 256), 256, 0, stream>>>(
                e_src[ei], e_dst[ei], alpha, den, vrel, outb[dt], NE);
        }
        for (int t = 0; t < 2; ++t) {
            int tb = lb + t * 9;
            long n = (long)Nn[t] * HID_;
            gelu_kernel<<<(unsigned)((n + 255) / 256), 256, 0, stream>>>(outb[t], gtmp, n);
            launch_gemm(4, gtmp, HID_, wt_main[(l * 2 + t) * 4 + 3], P(tb + 7), nxt[t],
                        HID_, Nn[t], HID_, HID_, 2, P(tb + 8), cur[t], HID_,
                        1, 0, 0, 0, stream);
        }
        for (int t = 0; t < 2; ++t) { float* tmp = cur[t]; cur[t] = nxt[t]; nxt[t] = tmp; }
    }

    // ---- final linears -> d_out (out_user then out_loc, concatenated) ----
    float* out_f = (float*)d_out;
    launch_gemm(4, cur[0], HID_, wt_lin[0], P(53), out_f, HID_,
                NU, HID_, HID_, 0, nullptr, nullptr, 0, 1, 0, 0, 0, stream);
    launch_gemm(4, cur[1], HID_, wt_lin[1], P(55), out_f + (size_t)NU * HID_, HID_,
                NL, HID_, HID_, 0, nullptr, nullptr, 0, 1, 0, 0, 0, stream);
}